// IPA_10445360464048
// MI455X (gfx1250) — compile-verified
//
#include <hip/hip_runtime.h>
#include <hip/hip_bf16.h>
#include <math.h>

typedef float v2f __attribute__((ext_vector_type(2)));
typedef float v8f __attribute__((ext_vector_type(8)));

#define N_     512
#define CS_    384
#define CZ_    128
#define CH_    16
#define H_     12
#define PQ_    4
#define PV_    8
#define FEAT_  960      // H_ * COD (COD = 32+16+32 = 80)
#define PROJW_ 1152     // 192(q) + 384(kv) + 144(qp) + 432(kvp)
#define NN_    (N_*N_)  // 262144

// D = A(16x4 f32) * B(4x16 f32) + C, full-precision WMMA (CDNA5 V_WMMA_F32_16X16X4_F32)
__device__ __forceinline__ v8f wmma4(v2f a, v2f b, v8f c) {
  return __builtin_amdgcn_wmma_f32_16x16x4_f32(false, a, false, b, (short)0, c, false, false);
}

// ---------------------------------------------------------------------------
// Kernel 1: all s-projections as one 512x1152 GEMM: proj = s @ [Wq|Wkv|Wqp|Wkvp]^T + bias
// grid (72 col-tiles, 32 row-tiles), 32 threads (1 wave) per block
// ---------------------------------------------------------------------------
__global__ __launch_bounds__(32) void k_proj(
    const float* __restrict__ s,
    const float* __restrict__ w_q,  const float* __restrict__ b_q,
    const float* __restrict__ w_kv, const float* __restrict__ b_kv,
    const float* __restrict__ w_qp, const float* __restrict__ b_qp,
    const float* __restrict__ w_kvp,const float* __restrict__ b_kvp,
    float* __restrict__ proj) {
  int lane = threadIdx.x;
  int it = blockIdx.y;             // i tile
  int ct = blockIdx.x;             // col tile
  int col0 = ct * 16;
  const float* w; const float* bias; int segoff;
  if (col0 < 192)      { w = w_q;   bias = b_q;   segoff = 0;   }
  else if (col0 < 576) { w = w_kv;  bias = b_kv;  segoff = 192; }
  else if (col0 < 720) { w = w_qp;  bias = b_qp;  segoff = 576; }
  else                 { w = w_kvp; bias = b_kvp; segoff = 720; }
  int m  = lane & 15;
  int kh = (lane >> 4) << 1;
  int hi8 = (lane >> 4) << 3;
  const float* arow = s + (it*16 + m) * CS_;                 // A[m][k] = s[i][k]
  const float* brow = w + (col0 - segoff + m) * CS_;         // B[k][n] = w[n][k]
  v8f acc = {};
  for (int k = 0; k < CS_; k += 4) {
    v2f a = { arow[k+kh], arow[k+kh+1] };
    v2f b = { brow[k+kh], brow[k+kh+1] };
    acc = wmma4(a, b, acc);
  }
  #pragma unroll
  for (int r = 0; r < 8; ++r) {
    int i = it*16 + r + hi8;
    int c = col0 + m;
    proj[i * PROJW_ + c] = acc[r] + bias[c - segoff];
  }
}

// ---------------------------------------------------------------------------
// Kernel 2: apply frames to q/k/v points (global coords)
// ---------------------------------------------------------------------------
__global__ void k_points(const float* __restrict__ proj,
                         const float* __restrict__ rot, const float* __restrict__ trans,
                         float* __restrict__ q_pts, float* __restrict__ k_pts,
                         float* __restrict__ v_pts) {
  int tid = blockIdx.x * blockDim.x + threadIdx.x;
  if (tid >= N_ * 192) return;                 // 48 q-points + 144 kv-points per residue
  int i = tid / 192;
  int p = tid % 192;
  const float* R = rot + i*9;
  const float* T = trans + i*3;
  float x, y, z;
  if (p < 48) {
    const float* src = proj + i*PROJW_ + 576;  // raw layout (3, 48)
    x = src[p]; y = src[48 + p]; z = src[96 + p];
  } else {
    int pp = p - 48;
    const float* src = proj + i*PROJW_ + 720;  // raw layout (3, 144)
    x = src[pp]; y = src[144 + pp]; z = src[288 + pp];
  }
  float gx = R[0]*x + R[1]*y + R[2]*z + T[0];
  float gy = R[3]*x + R[4]*y + R[5]*z + T[1];
  float gz = R[6]*x + R[7]*y + R[8]*z + T[2];
  if (p < 48) {
    float* dst = q_pts + (i*48 + p)*3;         // [i][h][pq][3], p = h*PQ+pq
    dst[0]=gx; dst[1]=gy; dst[2]=gz;
  } else {
    int pp = p - 48; int h = pp / 12; int pt = pp % 12;
    if (pt < PQ_) { float* d = k_pts + ((i*H_ + h)*PQ_ + pt)*3;       d[0]=gx; d[1]=gy; d[2]=gz; }
    else          { float* d = v_pts + ((i*H_ + h)*PV_ + (pt-PQ_))*3; d[0]=gx; d[1]=gy; d[2]=gz; }
  }
}

// ---------------------------------------------------------------------------
// Kernel 3: pair bias, first streaming pass over z (134 MB, HBM-bound)
// b_ws[h][ij] = sqrt(1/3) * (z[ij] . w_b[h] + b_b[h]); WMMA M=ij rows, K=128, N=16(12 valid)
// ---------------------------------------------------------------------------
__global__ __launch_bounds__(32) void k_bias(const float* __restrict__ z,
                                             const float* __restrict__ w_b,
                                             const float* __restrict__ b_b,
                                             float* __restrict__ b_ws) {
  int lane = threadIdx.x;
  int row0 = blockIdx.x * 16;                  // ij tile
  int m  = lane & 15;
  int kh = (lane >> 4) << 1;
  int hi8 = (lane >> 4) << 3;
  const float* arow = z + (long long)(row0 + m) * CZ_;
  float bg = (m < H_) ? 1.f : 0.f;             // branch-free column guard (EXEC stays all-1s)
  const float* brow = w_b + (m < H_ ? m : 0) * CZ_;
  v8f acc = {};
  for (int k = 0; k < CZ_; k += 4) {
    v2f a = { arow[k+kh], arow[k+kh+1] };
    v2f b = { brow[k+kh]*bg, brow[k+kh+1]*bg };
    acc = wmma4(a, b, acc);
  }
  const float s13 = 0.57735026918962576f;      // sqrt(1/3)
  if (m < H_) {
    #pragma unroll
    for (int r = 0; r < 8; ++r) {
      long long row = row0 + r + hi8;
      b_ws[(long long)m * NN_ + row] = (acc[r] + b_b[m]) * s13;
    }
  }
}

// ---------------------------------------------------------------------------
// Kernel 4: fused attention core, one wave per (head, 16-row i-tile)
// logits (qk WMMA + bias + point-attn + mask) -> softmax in LDS -> a@v, a@v_pts WMMA
// ---------------------------------------------------------------------------
__global__ __launch_bounds__(32) void k_attn(
    const float* __restrict__ proj,
    const float* __restrict__ q_pts, const float* __restrict__ k_pts,
    const float* __restrict__ v_pts,
    const float* __restrict__ b_ws,  const float* __restrict__ mask,
    const float* __restrict__ head_weights,
    float* __restrict__ a_ws, float* __restrict__ feats, float* __restrict__ opt_raw) {
  __shared__ float lg[16][N_];                 // 32 KB logits/probs (WGP has 320 KB LDS)
  __shared__ float qp[16][12];
  __shared__ float kp[16][12];
  int lane = threadIdx.x;
  int h  = blockIdx.x >> 5;
  int it = blockIdx.x & 31;
  int i0 = it * 16;
  int m   = lane & 15;
  int kh  = (lane >> 4) << 1;
  int hi8 = (lane >> 4) << 3;

  float hw = log1pf(__expf(head_weights[h])) * 0.13608276348795434f; // sqrt(1/54)
  const float qscale = 0.14433756729740643f;                         // sqrt(1/(3*16))

  for (int t = lane; t < 192; t += 32) {       // q-points for this row tile
    int rr = t / 12, e = t % 12;
    qp[rr][e] = q_pts[(i0 + rr)*144 + h*12 + e];
  }

  // ---- pass 1: logits ----
  const float* qrow = proj + (i0 + m)*PROJW_ + h*CH_;        // A[m][c] = q[i][h][c]
  for (int jt = 0; jt < 32; ++jt) {
    int j0 = jt * 16;
    __syncthreads();
    for (int t = lane; t < 192; t += 32) {
      int rr = t / 12, e = t % 12;
      kp[rr][e] = k_pts[(j0 + rr)*144 + h*12 + e];
    }
    __syncthreads();
    const float* krow = proj + (j0 + m)*PROJW_ + 192 + h*32; // B[c][j] = k[j][h][c]
    v8f acc = {};
    for (int k = 0; k < CH_; k += 4) {
      v2f a = { qrow[k+kh], qrow[k+kh+1] };
      v2f b = { krow[k+kh], krow[k+kh+1] };
      acc = wmma4(a, b, acc);
    }
    int j = j0 + m;
    float mj = mask[j];
    #pragma unroll
    for (int r = 0; r < 8; ++r) {
      int il = r + hi8;
      int i  = i0 + il;
      float logit = acc[r] * qscale;
      logit += b_ws[(long long)h * NN_ + (long long)i * N_ + j]; // pre-scaled by sqrt(1/3)
      float sq = 0.f;
      #pragma unroll
      for (int e = 0; e < 12; ++e) { float d = qp[il][e] - kp[m][e]; sq += d*d; }
      logit -= 0.5f * hw * sq;
      logit += 100000.0f * (mask[i]*mj - 1.0f);
      lg[il][j] = logit;
    }
  }
  __syncthreads();

  // ---- pass 2: softmax per row (wave32 shuffle reductions) + spill a ----
  for (int rr = 0; rr < 16; ++rr) {
    float mx = -1e30f;
    for (int j = lane; j < N_; j += 32) mx = fmaxf(mx, lg[rr][j]);
    for (int off = 16; off > 0; off >>= 1) mx = fmaxf(mx, __shfl_xor(mx, off, 32));
    float sum = 0.f;
    for (int j = lane; j < N_; j += 32) { float e = __expf(lg[rr][j] - mx); lg[rr][j] = e; sum += e; }
    for (int off = 16; off > 0; off >>= 1) sum += __shfl_xor(sum, off, 32);
    float inv = 1.0f / sum;
    for (int j = lane; j < N_; j += 32) {
      float p = lg[rr][j] * inv;
      lg[rr][j] = p;
      a_ws[(long long)h * NN_ + (long long)(i0 + rr) * N_ + j] = p;
    }
  }
  __syncthreads();

  // ---- pass 3: o = a@v (16 cols), o_pt = a@v_pts (24 cols -> two tiles) ----
  v8f acc_o = {}, acc_p0 = {}, acc_p1 = {};
  int vcol = 192 + h*32 + 16 + m;              // v at proj col 192 + h*32 + 16
  int pcol = h*24;                             // v_pts row layout: i*288 + h*24 + (p*3+d)
  int n1   = (m + 16 < 24) ? (16 + m) : 0;
  float g1 = (m + 16 < 24) ? 1.f : 0.f;
  for (int jt = 0; jt < 32; ++jt) {
    int j0 = jt * 16;
    for (int k = 0; k < 16; k += 4) {
      int j = j0 + k + kh;
      v2f a   = { lg[m][j], lg[m][j+1] };                          // A[i][j]
      v2f bo  = { proj[j*PROJW_ + vcol], proj[(j+1)*PROJW_ + vcol] };
      v2f bp0 = { v_pts[j*288 + pcol + m],  v_pts[(j+1)*288 + pcol + m] };
      v2f bp1 = { v_pts[j*288 + pcol + n1]*g1, v_pts[(j+1)*288 + pcol + n1]*g1 };
      acc_o  = wmma4(a, bo,  acc_o);
      acc_p0 = wmma4(a, bp0, acc_p0);
      acc_p1 = wmma4(a, bp1, acc_p1);
    }
  }
  #pragma unroll
  for (int r = 0; r < 8; ++r) {
    int i = i0 + r + hi8;
    feats[i*FEAT_ + h*CH_ + m] = acc_o[r];
    opt_raw[i*288 + pcol + m] = acc_p0[r];
    if (m + 16 < 24) opt_raw[i*288 + pcol + 16 + m] = acc_p1[r];
  }
}

// ---------------------------------------------------------------------------
// Kernel 5: inverse frame transform of o_pt + distances -> feats sections
// ---------------------------------------------------------------------------
__global__ void k_opt(const float* __restrict__ opt_raw, const float* __restrict__ rot,
                      const float* __restrict__ trans, float* __restrict__ feats) {
  int tid = blockIdx.x * blockDim.x + threadIdx.x;
  if (tid >= N_ * H_ * PV_) return;
  int i  = tid / 96;
  int hp = tid % 96;                           // h*PV + p
  const float* R = rot + i*9;
  const float* T = trans + i*3;
  const float* src = opt_raw + i*288 + hp*3;
  float x = src[0]-T[0], y = src[1]-T[1], z = src[2]-T[2];
  float lx = R[0]*x + R[3]*y + R[6]*z;         // R^T v
  float ly = R[1]*x + R[4]*y + R[7]*z;
  float lz = R[2]*x + R[5]*y + R[8]*z;
  float d = sqrtf(lx*lx + ly*ly + lz*lz + 1e-5f);
  float* f = feats + i*FEAT_;
  f[192 + hp] = lx; f[288 + hp] = ly; f[384 + hp] = lz; f[480 + hp] = d;
}

// ---------------------------------------------------------------------------
// Kernel 6: second z pass (fused pair output). Per residue i (1 wave):
//   t = a_i(12x512) @ z_i(512x128) via WMMA (A-frag reused across 8 col-tiles,
//   z row read exactly once), then o_pair = t @ w_dz^T through LDS.
// ---------------------------------------------------------------------------
__global__ __launch_bounds__(32) void k_pair(const float* __restrict__ z,
                                             const float* __restrict__ a_ws,
                                             const float* __restrict__ w_dz,
                                             const float* __restrict__ b_dz,
                                             float* __restrict__ feats) {
  __shared__ float tsh[16][CZ_];               // 8 KB
  int lane = threadIdx.x;
  int i = blockIdx.x;
  int m   = lane & 15;
  int kh  = (lane >> 4) << 1;
  int hi8 = (lane >> 4) << 3;
  float ag = (m < H_) ? 1.f : 0.f;
  const float* arow = a_ws + (long long)(m < H_ ? m : 0) * NN_ + (long long)i * N_; // A[h][j]
  const float* zrow = z + (long long)i * N_ * CZ_;
  v8f acc[8] = {};
  for (int k = 0; k < N_; k += 4) {
    int j = k + kh;
    v2f a = { arow[j]*ag, arow[j+1]*ag };
    const float* z0 = zrow + j*CZ_ + m;
    const float* z1 = z0 + CZ_;
    #pragma unroll
    for (int ct = 0; ct < 8; ++ct) {
      v2f b = { z0[ct*16], z1[ct*16] };        // B[j][c]
      acc[ct] = wmma4(a, b, acc[ct]);
    }
  }
  #pragma unroll
  for (int ct = 0; ct < 8; ++ct)
    #pragma unroll
    for (int r = 0; r < 8; ++r)
      tsh[r + hi8][ct*16 + m] = acc[ct][r];
  __syncthreads();
  for (int oidx = lane; oidx < H_*32; oidx += 32) {
    int hh = oidx >> 5, d = oidx & 31;
    const float* wrow = w_dz + d*CZ_;
    float sum = b_dz[d];
    for (int c = 0; c < CZ_; ++c) sum += tsh[hh][c] * wrow[c];
    feats[i*FEAT_ + 576 + hh*32 + d] = sum;
  }
}

// ---------------------------------------------------------------------------
// Kernel 7: out = feats(512x960) @ w_out^T(960x384) + b_out, WMMA GEMM
// ---------------------------------------------------------------------------
__global__ __launch_bounds__(32) void k_out(const float* __restrict__ feats,
                                            const float* __restrict__ w_out,
                                            const float* __restrict__ b_out,
                                            float* __restrict__ out) {
  int lane = threadIdx.x;
  int it = blockIdx.y;                         // 0..31
  int nt = blockIdx.x;                         // 0..23
  int m   = lane & 15;
  int kh  = (lane >> 4) << 1;
  int hi8 = (lane >> 4) << 3;
  const float* arow = feats + (it*16 + m) * FEAT_;
  const float* brow = w_out + (nt*16 + m) * FEAT_;
  v8f acc = {};
  for (int k = 0; k < FEAT_; k += 4) {
    v2f a = { arow[k+kh], arow[k+kh+1] };
    v2f b = { brow[k+kh], brow[k+kh+1] };
    acc = wmma4(a, b, acc);
  }
  #pragma unroll
  for (int r = 0; r < 8; ++r) {
    int i = it*16 + r + hi8;
    int c = nt*16 + m;
    out[i*CS_ + c] = acc[r] + b_out[c];
  }
}

// ---------------------------------------------------------------------------
extern "C" void kernel_launch(void* const* d_in, const int* in_sizes, int n_in,
                              void* d_out, int out_size, void* d_ws, size_t ws_size,
                              hipStream_t stream) {
  const float* s     = (const float*)d_in[0];
  const float* z     = (const float*)d_in[1];
  const float* rot   = (const float*)d_in[2];
  const float* trans = (const float*)d_in[3];
  const float* mask  = (const float*)d_in[4];
  const float* w_q   = (const float*)d_in[5];
  const float* b_q   = (const float*)d_in[6];
  const float* w_kv  = (const float*)d_in[7];
  const float* b_kv  = (const float*)d_in[8];
  const float* w_qp  = (const float*)d_in[9];
  const float* b_qp  = (const float*)d_in[10];
  const float* w_kvp = (const float*)d_in[11];
  const float* b_kvp = (const float*)d_in[12];
  const float* w_b   = (const float*)d_in[13];
  const float* b_b   = (const float*)d_in[14];
  const float* w_dz  = (const float*)d_in[15];
  const float* b_dz  = (const float*)d_in[16];
  const float* w_out = (const float*)d_in[17];
  const float* b_out = (const float*)d_in[18];
  const float* hwts  = (const float*)d_in[19];
  float* out = (float*)d_out;

  float* ws = (float*)d_ws;
  float* proj    = ws;                       // 512*1152   = 589824
  float* q_pts   = proj    + 589824;         // 512*144    = 73728
  float* k_pts   = q_pts   + 73728;          // 73728
  float* v_pts   = k_pts   + 73728;          // 512*288    = 147456
  float* b_ws    = v_pts   + 147456;         // 12*262144  = 3145728
  float* a_ws    = b_ws    + 3145728;        // 3145728
  float* opt_raw = a_ws    + 3145728;        // 512*288    = 147456
  float* feats   = opt_raw + 147456;         // 512*960    = 491520
  // total ~31.3 MB of scratch

  k_proj  <<<dim3(72, 32), 32, 0, stream>>>(s, w_q, b_q, w_kv, b_kv, w_qp, b_qp,
                                            w_kvp, b_kvp, proj);
  k_points<<<dim3(384), 256, 0, stream>>>(proj, rot, trans, q_pts, k_pts, v_pts);
  k_bias  <<<dim3(NN_/16), 32, 0, stream>>>(z, w_b, b_b, b_ws);
  k_attn  <<<dim3(H_*32), 32, 0, stream>>>(proj, q_pts, k_pts, v_pts, b_ws, mask,
                                           hwts, a_ws, feats, opt_raw);
  k_opt   <<<dim3(192), 256, 0, stream>>>(opt_raw, rot, trans, feats);
  k_pair  <<<dim3(N_), 32, 0, stream>>>(z, a_ws, w_dz, b_dz, feats);
  k_out   <<<dim3(24, 32), 32, 0, stream>>>(feats, w_out, b_out, out);
}